// custom_v7_180388626856
// MI455X (gfx1250) — compile-verified
//
#include <hip/hip_runtime.h>

typedef __attribute__((ext_vector_type(2))) float v2f;
typedef __attribute__((ext_vector_type(4))) float v4f;
typedef __attribute__((ext_vector_type(8))) float v8f;

#define B_DIM   64      // batch
#define T_DIM   64      // reduction dim
#define U_DIM   64      // u*u outputs per pixel
#define RECON   512
#define LROW    68      // padded LDS row stride (dwords): conflict-free + 8B-aligned b64 frags

// One workgroup (256 threads = 8 wave32) per pixel p in [0,4096).
// C[p] (64x64) = Y[p] (64x64) @ W[p] (64x64) + bias[p], via v_wmma_f32_16x16x4_f32.
__global__ __launch_bounds__(256) void pixel_gemm_wmma_f32(
    const float* __restrict__ yt,    // [64, 64, 4096]  = [b][t][p]
    const float* __restrict__ Ht,    // [64, 512, 512]  = [t][r][c]
    const float* __restrict__ bias,  // [4096, 64]      = [p][u]
    float* __restrict__ out)         // [64, 512, 512]  = [b][r][c]
{
    __shared__ float Ylds[B_DIM * LROW];   // [b][t], padded
    __shared__ float Wt[U_DIM * LROW];     // [u][t] (transposed!), padded

    const int p   = blockIdx.x;            // pixel index
    const int iy  = p >> 6;
    const int ix  = p & 63;
    const int tid = threadIdx.x;

    // ---- Stage Y[p]: Ylds[b][t] = yt[(b*64+t)*4096 + p] (L2-amortized stride gather) ----
    {
        const float* src = yt + p;
        #pragma unroll
        for (int i = 0; i < 16; ++i) {
            const int idx = tid + i * 256;                 // idx = b*64 + t
            const float v = src[(size_t)idx * 4096];
            Ylds[(idx >> 6) * LROW + (idx & 63)] = v;      // consecutive banks per wave
        }
    }

    // ---- Stage W[p] transposed: Wt[uy*8+ux][t] = Ht[t][iy*8+uy][ix*8+ux] ----
    // chunk -> (t = cidx&63, uy = cidx>>6): per-wave t varies => conflict-free LDS stores.
    {
        const float* hbase = Ht + (size_t)(iy * 8) * RECON + ix * 8;
        #pragma unroll
        for (int i = 0; i < 2; ++i) {
            const int cidx = tid + i * 256;                // 0..511
            const int t    = cidx & 63;
            const int uy   = cidx >> 6;                    // 0..7
            const float* src = hbase + (size_t)t * (RECON * RECON) + uy * RECON;
            const v4f lo = *(const v4f*)(src);             // 32B contiguous global chunk
            const v4f hi = *(const v4f*)(src + 4);
            #pragma unroll
            for (int ux = 0; ux < 4; ++ux)
                Wt[(uy * 8 + ux) * LROW + t] = lo[ux];
            #pragma unroll
            for (int ux = 0; ux < 4; ++ux)
                Wt[(uy * 8 + 4 + ux) * LROW + t] = hi[ux];
        }
    }
    __syncthreads();

    // ---- WMMA compute: 8 waves, 2 adjacent 16x16 N-tiles per wave (shared A) ----
    const int lane = tid & 31;
    const int wave = tid >> 5;
    const int hi   = lane >> 4;          // lane half select
    const int ln   = lane & 15;

    const int mt   = wave >> 1;          // row tile 0..3
    const int nt0  = (wave & 1) * 2;     // col tile base: 0 or 2
    const int col0 = nt0 * 16 + ln;      // u column for tile 0
    const int col1 = col0 + 16;          // u column for tile 1

    v8f c0 = {};
    v8f c1 = {};
    const float* yrow  = &Ylds[(mt * 16 + ln) * LROW];  // A: row m = ln
    const float* wrow0 = &Wt[col0 * LROW];              // B tile 0: column n = col0
    const float* wrow1 = &Wt[col1 * LROW];              // B tile 1: column n = col1

    #pragma unroll
    for (int k0 = 0; k0 < T_DIM; k0 += 4) {
        const int kb = k0 + 2 * hi;      // ISA 16x16x4 f32 layout: K = j + 2*(lane>>4); kb even
        const v2f a  = *(const v2f*)(yrow  + kb);   // single aligned ds_load_b64
        const v2f b0 = *(const v2f*)(wrow0 + kb);
        const v2f b1 = *(const v2f*)(wrow1 + kb);
        // 8 args: (neg_a, A, neg_b, B, c_mod, C, reuse_a, reuse_b)
        c0 = __builtin_amdgcn_wmma_f32_16x16x4_f32(false, a, false, b0, (short)0, c0, false, false);
        c1 = __builtin_amdgcn_wmma_f32_16x16x4_f32(false, a, false, b1, (short)0, c1, false, false);
    }

    // ---- Epilogue: add bias[p][u], pixel-shuffle store ----
    const float bias0 = bias[p * U_DIM + col0];
    const float bias1 = bias[p * U_DIM + col1];
    const int uy0 = col0 >> 3, ux0 = col0 & 7;
    const int uy1 = col1 >> 3, ux1 = col1 & 7;
    const int roff0 = (iy * 8 + uy0) * RECON + ix * 8 + ux0;
    const int roff1 = (iy * 8 + uy1) * RECON + ix * 8 + ux1;

    #pragma unroll
    for (int r = 0; r < 8; ++r) {
        const int b = mt * 16 + hi * 8 + r;          // C layout: M = r + 8*(lane>>4)
        const size_t base = (size_t)b * (RECON * RECON);
        out[base + roff0] = c0[r] + bias0;
        out[base + roff1] = c1[r] + bias1;
    }
}

extern "C" void kernel_launch(void* const* d_in, const int* in_sizes, int n_in,
                              void* d_out, int out_size, void* d_ws, size_t ws_size,
                              hipStream_t stream) {
    (void)in_sizes; (void)n_in; (void)out_size; (void)d_ws; (void)ws_size;
    const float* yt     = (const float*)d_in[0];   // [64,64,64,64]
    const float* Ht     = (const float*)d_in[1];   // [64,512,512]
    const float* biases = (const float*)d_in[2];   // [4096,1,64]
    float* out          = (float*)d_out;           // [64,1,512,512]

    dim3 grid(4096);
    dim3 block(256);
    hipLaunchKernelGGL(pixel_gemm_wmma_f32, grid, block, 0, stream, yt, Ht, biases, out);
}